// OurLayer_GAT_79800492359726
// MI455X (gfx1250) — compile-verified
//
#include <hip/hip_runtime.h>
#include <hip/hip_bf16.h>
#include <math.h>

typedef __attribute__((ext_vector_type(2))) float v2f;
typedef __attribute__((ext_vector_type(8))) float v8f;

#define GAT_H 3
#define GAT_D 16
#define GAT_F 128
#define GAT_HD 48
#define NEG_SLOPE 0.2f
// ordered-int encoding of -inf (0xFF800000 ^ 0x7FFFFFFF)
#define ENC_NEG_INF ((int)0x807FFFFF)

// ---------------------------------------------------------------------------
// 1) feat = x @ W^T via V_WMMA_F32_16X16X4_F32. One wave computes a 16-row
//    tile across all 48 output columns (3 heads x 16). W (48x128, 24KB) is
//    staged in LDS with a padded stride of 130 floats so the 16 lanes of each
//    half-wave hit distinct banks on their float2 B-fragment loads.
// ---------------------------------------------------------------------------
__global__ void gat_gemm_wmma(const float* __restrict__ x,
                              const float* __restrict__ W,
                              float* __restrict__ feat, int N) {
  __shared__ float wlds[GAT_HD * 130];
  for (int i = threadIdx.x; i < GAT_HD * GAT_F; i += blockDim.x) {
    int r = i >> 7, c = i & 127;
    wlds[r * 130 + c] = W[r * GAT_F + c];
  }
  __syncthreads();

  const int wave = threadIdx.x >> 5;          // 8 waves / block
  const int lane = threadIdx.x & 31;
  const int tile = blockIdx.x * 8 + wave;     // 16-row output tile
  if (tile * 16 >= N) return;                 // whole-wave uniform exit
  const int half = lane >> 4;                 // lane half selects K pair
  const int l16  = lane & 15;

  int row = tile * 16 + l16;
  int rowc = row < N ? row : N - 1;           // clamp tail reads
  const float* xrow = x + (size_t)rowc * GAT_F;

  v8f acc0 = {}, acc1 = {}, acc2 = {};
  for (int k = 0; k < GAT_F; k += 4) {
    const int ka = k + 2 * half;
    // A fragment: lane l16 = row M, VGPR0/1 hold K = ka, ka+1
    v2f a = *(const v2f*)(xrow + ka);
    // B fragments: lane l16 = col N within each head's 16-col group
    v2f b0 = *(const v2f*)(&wlds[(0 * 16 + l16) * 130 + ka]);
    v2f b1 = *(const v2f*)(&wlds[(1 * 16 + l16) * 130 + ka]);
    v2f b2 = *(const v2f*)(&wlds[(2 * 16 + l16) * 130 + ka]);
    acc0 = __builtin_amdgcn_wmma_f32_16x16x4_f32(false, a, false, b0,
                                                 (short)0, acc0, false, false);
    acc1 = __builtin_amdgcn_wmma_f32_16x16x4_f32(false, a, false, b1,
                                                 (short)0, acc1, false, false);
    acc2 = __builtin_amdgcn_wmma_f32_16x16x4_f32(false, a, false, b2,
                                                 (short)0, acc2, false, false);
  }
  // D layout: VGPR i holds M = i (lanes 0-15) / M = i+8 (lanes 16-31), N = l16
  for (int i = 0; i < 8; ++i) {
    int r = tile * 16 + i + 8 * half;
    if (r < N) {
      float* fr = feat + (size_t)r * GAT_HD;
      fr[0 * 16 + l16] = acc0[i];
      fr[1 * 16 + l16] = acc1[i];
      fr[2 * 16 + l16] = acc2[i];
    }
  }
}

// ---------------------------------------------------------------------------
// 2) per-(node,head) attention logits el/er
// ---------------------------------------------------------------------------
__global__ void gat_elr(const float* __restrict__ feat,
                        const float* __restrict__ attn_l,
                        const float* __restrict__ attn_r,
                        float* __restrict__ el, float* __restrict__ er, int NH) {
  int i = blockIdx.x * blockDim.x + threadIdx.x;
  if (i >= NH) return;
  int n = i / GAT_H, h = i % GAT_H;
  const float* f  = feat + (size_t)n * GAT_HD + h * GAT_D;
  const float* al = attn_l + h * GAT_D;
  const float* ar = attn_r + h * GAT_D;
  float sl = 0.f, sr = 0.f;
#pragma unroll
  for (int d = 0; d < GAT_D; ++d) { sl += f[d] * al[d]; sr += f[d] * ar[d]; }
  el[i] = sl; er[i] = sr;
}

// ---------------------------------------------------------------------------
// 3) init: out accumulator = 0, emax = enc(-inf), denom = 0
// ---------------------------------------------------------------------------
__global__ void gat_init(int* __restrict__ emax_bits, float* __restrict__ denom,
                         float* __restrict__ out, int NH, int NF) {
  int i = blockIdx.x * blockDim.x + threadIdx.x;
  if (i < NF) out[i] = 0.0f;
  if (i < NH) { emax_bits[i] = ENC_NEG_INF; denom[i] = 0.0f; }
}

__device__ __forceinline__ float leaky(float e) {
  return e > 0.0f ? e : NEG_SLOPE * e;
}
__device__ __forceinline__ int enc_f32(float f) {
  int b = __float_as_int(f);
  return b >= 0 ? b : (b ^ 0x7fffffff);
}

// ---------------------------------------------------------------------------
// 4) segment max over dst via ordered-int atomicMax
// ---------------------------------------------------------------------------
__global__ void gat_edge_max(const int* __restrict__ src, const int* __restrict__ dst,
                             const float* __restrict__ el, const float* __restrict__ er,
                             int* __restrict__ emax_bits, int E) {
  int i = blockIdx.x * blockDim.x + threadIdx.x;
  if (i >= E) return;
  int s = src[i] * GAT_H, d = dst[i] * GAT_H;
#pragma unroll
  for (int h = 0; h < GAT_H; ++h) {
    float e = leaky(el[s + h] + er[d + h]);
    atomicMax(&emax_bits[d + h], enc_f32(e));
  }
}

// 5) decode emax in place; non-finite (no in-edges) -> 0
__global__ void gat_emax_fin(int* __restrict__ emax_bits,
                             float* __restrict__ emax_f, int NH) {
  int i = blockIdx.x * blockDim.x + threadIdx.x;
  if (i >= NH) return;
  int b = emax_bits[i];
  b = b >= 0 ? b : (b ^ 0x7fffffff);
  float f = __int_as_float(b);
  if (!isfinite(f)) f = 0.0f;
  emax_f[i] = f;
}

// 6) denom = segment_sum(exp(e - emax[dst]))
__global__ void gat_edge_den(const int* __restrict__ src, const int* __restrict__ dst,
                             const float* __restrict__ el, const float* __restrict__ er,
                             const float* __restrict__ emax, float* __restrict__ denom,
                             int E) {
  int i = blockIdx.x * blockDim.x + threadIdx.x;
  if (i >= E) return;
  int s = src[i] * GAT_H, d = dst[i] * GAT_H;
#pragma unroll
  for (int h = 0; h < GAT_H; ++h) {
    float e = leaky(el[s + h] + er[d + h]);
    atomicAdd(&denom[d + h], __expf(e - emax[d + h]));
  }
}

// 7) out[dst] += feat[src] * a   (msg accumulation; feat is L2-resident)
__global__ void gat_edge_msg(const int* __restrict__ src, const int* __restrict__ dst,
                             const float* __restrict__ el, const float* __restrict__ er,
                             const float* __restrict__ emax, const float* __restrict__ denom,
                             const float* __restrict__ feat, float* __restrict__ out,
                             int EH) {
  int i = blockIdx.x * blockDim.x + threadIdx.x;
  if (i >= EH) return;
  int e = i / GAT_H, h = i % GAT_H;
  int s = src[e], d = dst[e];
  float t = leaky(el[s * GAT_H + h] + er[d * GAT_H + h]);
  float a = __expf(t - emax[d * GAT_H + h]) / denom[d * GAT_H + h];
  const float* fs = feat + (size_t)s * GAT_HD + h * GAT_D;
  float* od = out + (size_t)d * GAT_HD + h * GAT_D;
#pragma unroll
  for (int dd = 0; dd < GAT_D; ++dd) atomicAdd(&od[dd], fs[dd] * a);
}

// 8) out = (1-lin)*msg + lin*feat
__global__ void gat_final(const float* __restrict__ feat, const float* __restrict__ lin,
                          float* __restrict__ out, int NF) {
  int i = blockIdx.x * blockDim.x + threadIdx.x;
  if (i >= NF) return;
  float l = lin[i / GAT_HD];
  out[i] = (1.0f - l) * out[i] + l * feat[i];
}

extern "C" void kernel_launch(void* const* d_in, const int* in_sizes, int n_in,
                              void* d_out, int out_size, void* d_ws, size_t ws_size,
                              hipStream_t stream) {
  const float* x      = (const float*)d_in[0];
  const float* W      = (const float*)d_in[1];
  const float* attn_l = (const float*)d_in[2];
  const float* attn_r = (const float*)d_in[3];
  const float* lin    = (const float*)d_in[4];
  const int*   src    = (const int*)d_in[5];
  const int*   dst    = (const int*)d_in[6];
  float* out = (float*)d_out;

  const int N  = in_sizes[4];          // linear_comb has N elements
  const int E  = in_sizes[5];
  const int NH = N * GAT_H;
  const int NF = N * GAT_HD;
  const int EH = E * GAT_H;

  // workspace layout (~24 MB)
  char* p = (char*)d_ws;
  float* feat = (float*)p;      p += (size_t)NF * sizeof(float);
  float* el   = (float*)p;      p += (size_t)NH * sizeof(float);
  float* er   = (float*)p;      p += (size_t)NH * sizeof(float);
  int*   emax = (int*)p;        p += (size_t)NH * sizeof(int);   // bits then float
  float* den  = (float*)p;      p += (size_t)NH * sizeof(float);

  const int BLK = 256;
  int tiles = (N + 15) / 16;
  dim3 gGemm((tiles + 7) / 8);
  gat_gemm_wmma<<<gGemm, BLK, 0, stream>>>(x, W, feat, N);

  gat_elr<<<(NH + BLK - 1) / BLK, BLK, 0, stream>>>(feat, attn_l, attn_r, el, er, NH);
  gat_init<<<(NF + BLK - 1) / BLK, BLK, 0, stream>>>(emax, den, out, NH, NF);
  gat_edge_max<<<(E + BLK - 1) / BLK, BLK, 0, stream>>>(src, dst, el, er, emax, E);
  gat_emax_fin<<<(NH + BLK - 1) / BLK, BLK, 0, stream>>>(emax, (float*)emax, NH);
  gat_edge_den<<<(E + BLK - 1) / BLK, BLK, 0, stream>>>(src, dst, el, er, (float*)emax, den, E);
  gat_edge_msg<<<(EH + BLK - 1) / BLK, BLK, 0, stream>>>(src, dst, el, er, (float*)emax, den,
                                                         feat, out, EH);
  gat_final<<<(NF + BLK - 1) / BLK, BLK, 0, stream>>>(feat, lin, out, NF);
}